// Loss_20392504722119
// MI455X (gfx1250) — compile-verified
//
#include <hip/hip_runtime.h>

typedef __attribute__((ext_vector_type(2))) float v2f;
typedef __attribute__((ext_vector_type(8))) float v8f;

// ---------------------------------------------------------------------------
// Wave32 sum via V_WMMA_F32_16X16X4_F32:
//   A (16x4): lane l supplies VGPR0 -> A[l%16][l<16?0:2] = v, VGPR1 -> 0
//   B (4x16): all ones
//   D[m][n] = A[m][0]+A[m][1]+A[m][2]+A[m][3] = v_m + v_{m+16}   (any n)
// D layout: VGPR r, lanes 0-15 -> D[r][lane], lanes 16-31 -> D[r+8][lane-16].
// Summing the 8 D VGPRs gives q0..q7 (lanes 0-15) / q8..q15 (lanes 16-31);
// one shfl_xor(16) completes the full wave sum in every lane.
// ---------------------------------------------------------------------------
__device__ __forceinline__ float wave_sum_wmma(float v) {
  v2f a; a[0] = v;    a[1] = 0.0f;
  v2f b; b[0] = 1.0f; b[1] = 1.0f;
  v8f c = {};
  v8f d = __builtin_amdgcn_wmma_f32_16x16x4_f32(
      /*neg_a=*/false, a, /*neg_b=*/false, b,
      /*c_mod=*/(short)0, c, /*reuse_a=*/false, /*reuse_b=*/false);
  float s = ((d[0] + d[1]) + (d[2] + d[3])) + ((d[4] + d[5]) + (d[6] + d[7]));
  s += __shfl_xor(s, 16, 32);
  return s;  // full wave sum, all lanes
}

// Block-level (256 threads = 8 waves) deterministic sum. Result valid in wave 0.
__device__ __forceinline__ float block_sum(float v, float* lds) {
  const int lane = threadIdx.x & 31;
  const int wave = threadIdx.x >> 5;
  float w = wave_sum_wmma(v);
  if (lane == 0) lds[wave] = w;
  __syncthreads();
  float r = 0.0f;
  if (wave == 0) {  // uniform per wave -> EXEC stays all-ones inside
    float x = (lane < (int)(blockDim.x >> 5)) ? lds[lane] : 0.0f;
    r = wave_sum_wmma(x);
  }
  return r;
}

// S=7, B=2, C=20 -> pred row = 30 floats, target row = 25 floats per cell.
__global__ void yolo_loss_partial(const float* __restrict__ pred,
                                  const float* __restrict__ tgt,
                                  float* __restrict__ partial) {
  __shared__ float lds[8];
  const int cell = blockIdx.x * blockDim.x + threadIdx.x;  // grid sized exactly
  const float* p = pred + (size_t)cell * 30;
  const float* t = tgt  + (size_t)cell * 25;

  const int g  = cell % 49;
  const float gy = (float)(g / 7);
  const float gx = (float)(g % 7);

  const float tobj = t[0];
  const float tx = t[1], ty = t[2], tw = t[3], th = t[4];
  const float objf   = (tobj == 1.0f) ? 1.0f : 0.0f;
  const float noobjf = (tobj == 0.0f) ? 1.0f : 0.0f;

  // target box -> xyxy
  const float tcx = (gx + tx) / 7.0f;
  const float tcy = (gy + ty) / 7.0f;
  const float tx1 = tcx - tw * 0.5f, ty1 = tcy - th * 0.5f;
  const float tx2 = tcx + tw * 0.5f, ty2 = tcy + th * 0.5f;
  const float area_t = (tx2 - tx1) * (ty2 - ty1);

  float conf[2], bx[2], by[2], bw[2], bh[2], iou[2];
#pragma unroll
  for (int i = 0; i < 2; ++i) {
    conf[i] = p[i * 5 + 0];
    bx[i] = p[i * 5 + 1]; by[i] = p[i * 5 + 2];
    bw[i] = p[i * 5 + 3]; bh[i] = p[i * 5 + 4];
    const float ax = fabsf(bx[i]), ay = fabsf(by[i]);
    const float aw = fabsf(bw[i]), ah = fabsf(bh[i]);
    const float cx = (gx + ax) / 7.0f, cy = (gy + ay) / 7.0f;
    const float px1 = cx - aw * 0.5f, py1 = cy - ah * 0.5f;
    const float px2 = cx + aw * 0.5f, py2 = cy + ah * 0.5f;
    const float lx = fmaxf(px1, tx1), ly = fmaxf(py1, ty1);
    const float rx = fminf(px2, tx2), ry = fminf(py2, ty2);
    const float iw = fmaxf(rx - lx, 0.0f), ih = fmaxf(ry - ly, 0.0f);
    const float inter  = iw * ih;
    const float area_p = (px2 - px1) * (py2 - py1);
    iou[i] = inter / (area_p + area_t - inter);
  }

  // jnp.argmax picks the first max on ties -> best=1 only if strictly greater
  const int   best    = (iou[1] > iou[0]) ? 1 : 0;
  const float max_iou = best ? iou[1] : iou[0];
  const float rc = conf[best], rxv = bx[best], ryv = by[best];
  const float rwv = bw[best], rhv = bh[best];

  const float dx = rxv - tx, dy = ryv - ty;
  const float xy_loss = dx * dx + dy * dy;
  const float pw = sqrtf(fabsf(rwv)), ph = sqrtf(fabsf(rhv));
  const float tws = sqrtf(fabsf(tw)), ths = sqrtf(fabsf(th));
  const float dw = pw - tws, dh = ph - ths;
  const float wh_loss = dw * dw + dh * dh;
  const float dc = rc - max_iou;
  const float obj_conf = dc * dc;

  float cls = 0.0f;
#pragma unroll
  for (int k = 0; k < 20; ++k) {
    const float d = p[10 + k] - t[5 + k];
    cls += d * d;
  }

  const float loss =
      objf * (5.0f * (xy_loss + wh_loss) + obj_conf + cls) +
      0.5f * noobjf * (conf[0] * conf[0] + conf[1] * conf[1]);

  const float bsum = block_sum(loss, lds);
  if (threadIdx.x == 0) partial[blockIdx.x] = bsum;
}

__global__ void yolo_loss_final(const float* __restrict__ partial, int n,
                                float* __restrict__ out, float scale) {
  __shared__ float lds[8];
  float acc = 0.0f;
  for (int i = threadIdx.x; i < n; i += blockDim.x) acc += partial[i];
  const float s = block_sum(acc, lds);
  if (threadIdx.x == 0) out[0] = s * scale;
}

extern "C" void kernel_launch(void* const* d_in, const int* in_sizes, int n_in,
                              void* d_out, int out_size, void* d_ws, size_t ws_size,
                              hipStream_t stream) {
  (void)in_sizes; (void)n_in; (void)out_size; (void)ws_size;
  const float* pred = (const float*)d_in[0];  // (8192, 1470) fp32
  const float* tgt  = (const float*)d_in[1];  // (8192, 1225) fp32
  float* out     = (float*)d_out;
  float* partial = (float*)d_ws;              // 1568 floats of scratch

  const int M = 8192 * 49;        // 401408 cells
  const int threads = 256;        // 8 waves (wave32)
  const int blocks  = M / threads;  // 1568, exact -> no tail, EXEC all-ones

  yolo_loss_partial<<<blocks, threads, 0, stream>>>(pred, tgt, partial);
  yolo_loss_final<<<1, threads, 0, stream>>>(partial, blocks, out,
                                             1.0f / 8192.0f);
}